// LocalWindowAttention_42322607735296
// MI455X (gfx1250) — compile-verified
//
#include <hip/hip_runtime.h>
#include <hip/hip_bf16.h>

// ---------------------------------------------------------------------------
// LocalWindowAttention for MI455X (gfx1250, wave32, WMMA + async-to-LDS).
//   out  = (softmax_local(q k^T / 8) v) Wo^T + bo     q,k,v = x W{q,k,v}^T
//   d_out = [out (B*T*C)] ++ [attn (B*H*T*T)]   (reference returns a tuple)
// GEMMs: V_WMMA_F32_16X16X4_F32 (exact f32) with double-buffered LDS tiles
// filled by GLOBAL_LOAD_ASYNC_TO_LDS_B128 (ASYNCcnt-tracked, no VGPR bounce).
// End-to-end bound: the 1 GiB dense attn store (~46us @ 23.3 TB/s).
// ---------------------------------------------------------------------------

typedef __attribute__((ext_vector_type(2))) float v2f;
typedef __attribute__((ext_vector_type(4))) float v4f;
typedef __attribute__((ext_vector_type(8))) float v8f;

#define B_BATCH 4
#define T_SEQ   2048
#define C_DIM   1024
#define H_NUM   16
#define D_HEAD  64
#define W_HALF  4          // window half-width -> up to 9 keys

#define TILE_M 128
#define TILE_N 128
#define TILE_K 32
#define LDS_STRIDE 36      // 36*4B: 8B-aligned b64 reads, conflict-friendly pad

__device__ __forceinline__ void async_b128_to_lds(const float* gptr, unsigned lds_byte) {
    // LDS byte address = low 32 bits of the generic pointer (ISA: addr[31:0]).
    asm volatile("global_load_async_to_lds_b128 %0, %1, off"
                 :: "v"(lds_byte), "v"(gptr) : "memory");
}
__device__ __forceinline__ void wait_async0() {
    asm volatile("s_wait_asynccnt 0" ::: "memory");
}

// Y[M,N] = X[M,K] @ W[N,K]^T (+ bias[N])   — nn.Linear semantics.
__global__ __launch_bounds__(256)
void gemm_xt_wmma(const float* __restrict__ X,
                  const float* __restrict__ W,
                  const float* __restrict__ bias,
                  float* __restrict__ Y,
                  int M, int N, int K)
{
    __shared__ float Asm[2][TILE_M * LDS_STRIDE];
    __shared__ float Bsm[2][TILE_N * LDS_STRIDE];

    const int tid    = threadIdx.x;
    const int lane   = tid & 31;
    const int wave   = tid >> 5;          // 0..7
    const int wave_m = wave >> 2;         // 0..1 : 64 rows of the 128 tile
    const int wave_n = wave & 3;          // 0..3 : 32 cols of the 128 tile

    const int block_m = blockIdx.y * TILE_M;
    const int block_n = blockIdx.x * TILE_N;

    v8f acc[4][2];
    const v8f vzero = {0.f, 0.f, 0.f, 0.f, 0.f, 0.f, 0.f, 0.f};
#pragma unroll
    for (int i = 0; i < 4; ++i)
#pragma unroll
        for (int j = 0; j < 2; ++j)
            acc[i][j] = vzero;

    // Cooperative staging: each thread owns one b128 (16B) per row-pass.
    const int ld_row = tid >> 3;          // 0..31
    const int ld_col = (tid & 7) * 4;     // 0..28

    const int half = lane >> 4;           // K sub-offset select (ISA A layout)
    const int l15  = lane & 15;

    auto stage = [&](int buf, int k0) {
#pragma unroll
        for (int p = 0; p < 4; ++p) {
            const int r = ld_row + p * 32;
            async_b128_to_lds(X + (size_t)(block_m + r) * K + k0 + ld_col,
                              (unsigned)(uintptr_t)&Asm[buf][r * LDS_STRIDE + ld_col]);
            async_b128_to_lds(W + (size_t)(block_n + r) * K + k0 + ld_col,
                              (unsigned)(uintptr_t)&Bsm[buf][r * LDS_STRIDE + ld_col]);
        }
    };

    stage(0, 0);
    wait_async0();
    __syncthreads();

    int buf = 0;
    for (int k0 = 0; k0 < K; k0 += TILE_K) {
        if (k0 + TILE_K < K)              // prefetch next tile into other buffer
            stage(buf ^ 1, k0 + TILE_K);

        const float* Ab = &Asm[buf][0];
        const float* Bb = &Bsm[buf][0];
#pragma unroll
        for (int kk = 0; kk < TILE_K; kk += 4) {
            const int kcol = kk + half * 2;     // lanes 0-15: K,K+1; 16-31: K+2,K+3
            v2f afrag[4], bfrag[2];
#pragma unroll
            for (int i = 0; i < 4; ++i) {
                const int row = wave_m * 64 + i * 16 + l15;
                afrag[i] = *(const v2f*)&Ab[row * LDS_STRIDE + kcol];
            }
#pragma unroll
            for (int j = 0; j < 2; ++j) {
                const int row = wave_n * 32 + j * 16 + l15;
                bfrag[j] = *(const v2f*)&Bb[row * LDS_STRIDE + kcol];
            }
#pragma unroll
            for (int i = 0; i < 4; ++i)
#pragma unroll
                for (int j = 0; j < 2; ++j)
                    acc[i][j] = __builtin_amdgcn_wmma_f32_16x16x4_f32(
                        false, afrag[i], false, bfrag[j],
                        (short)0, acc[i][j], false, false);
        }

        wait_async0();      // my prefetch landed in LDS
        __syncthreads();    // everyone's landed; everyone done reading old buf
        buf ^= 1;
    }

    // C/D layout: VGPR r -> M=r (lanes 0-15), M=r+8 (lanes 16-31); N=lane&15.
#pragma unroll
    for (int i = 0; i < 4; ++i) {
        const int mbase = block_m + wave_m * 64 + i * 16 + half * 8;
#pragma unroll
        for (int j = 0; j < 2; ++j) {
            const int col = block_n + wave_n * 32 + j * 16 + l15;
            const float bv = bias ? bias[col] : 0.f;
#pragma unroll
            for (int r = 0; r < 8; ++r)
                Y[(size_t)(mbase + r) * N + col] = acc[i][j][r] + bv;
        }
    }
}

// Zero the 1 GiB dense attn tensor with streaming (non-temporal) stores.
__global__ __launch_bounds__(256)
void fill_attn_zero(v4f* __restrict__ dst, size_t n4)
{
    const v4f z = {0.f, 0.f, 0.f, 0.f};
    size_t i = (size_t)blockIdx.x * blockDim.x + threadIdx.x;
    const size_t stride = (size_t)gridDim.x * blockDim.x;
    for (; i < n4; i += stride)
        __builtin_nontemporal_store(z, dst + i);
}

// One wave32 per (b, h, t): 9-key windowed softmax-attention, D=64
// (each lane owns a contiguous float2 of the head dimension).
__global__ __launch_bounds__(256)
void local_attn(const float* __restrict__ q,
                const float* __restrict__ k,
                const float* __restrict__ v,
                float* __restrict__ y,
                float* __restrict__ attn)
{
    const int wave_global = blockIdx.x * 8 + (threadIdx.x >> 5);
    const int lane = threadIdx.x & 31;

    const int t  = wave_global % T_SEQ;
    const int bh = wave_global / T_SEQ;
    const int h  = bh % H_NUM;
    const int b  = bh / H_NUM;

    const size_t chan = (size_t)h * D_HEAD + lane * 2;
    const size_t qoff = ((size_t)(b * T_SEQ + t)) * C_DIM + chan;
    const float2 qv = *(const float2*)(q + qoff);

    const int j0 = (t - W_HALF < 0) ? 0 : t - W_HALF;
    const int j1 = (t + W_HALF >= T_SEQ) ? T_SEQ - 1 : t + W_HALF;
    const int nj = j1 - j0 + 1;                  // 5..9, wave-uniform

    float s[2 * W_HALF + 1];
#pragma unroll
    for (int idx = 0; idx < 2 * W_HALF + 1; ++idx) {
        if (idx < nj) {
            const size_t koff = ((size_t)(b * T_SEQ + j0 + idx)) * C_DIM + chan;
            const float2 kv = *(const float2*)(k + koff);
            float d = qv.x * kv.x + qv.y * kv.y;
#pragma unroll
            for (int off = 16; off; off >>= 1)
                d += __shfl_xor(d, off, 32);
            s[idx] = d * 0.125f;                 // 1/sqrt(64)
        } else {
            s[idx] = -__builtin_inff();
        }
    }

    float m = s[0];
#pragma unroll
    for (int idx = 1; idx < 2 * W_HALF + 1; ++idx) m = fmaxf(m, s[idx]);
    float p[2 * W_HALF + 1];
    float denom = 0.f;
#pragma unroll
    for (int idx = 0; idx < 2 * W_HALF + 1; ++idx) {
        p[idx] = (idx < nj) ? __expf(s[idx] - m) : 0.f;
        denom += p[idx];
    }
    const float inv = 1.f / denom;

    float2 acc; acc.x = 0.f; acc.y = 0.f;
    const size_t arow = (((size_t)(b * H_NUM + h)) * T_SEQ + t) * (size_t)T_SEQ;
#pragma unroll
    for (int idx = 0; idx < 2 * W_HALF + 1; ++idx) {
        if (idx < nj) {
            const float pj = p[idx] * inv;
            const size_t voff = ((size_t)(b * T_SEQ + j0 + idx)) * C_DIM + chan;
            const float2 vv = *(const float2*)(v + voff);
            acc.x += pj * vv.x;
            acc.y += pj * vv.y;
            if (lane == idx)
                __builtin_nontemporal_store(pj, attn + arow + j0 + idx);
        }
    }
    *(float2*)(y + qoff) = acc;
}

extern "C" void kernel_launch(void* const* d_in, const int* in_sizes, int n_in,
                              void* d_out, int out_size, void* d_ws, size_t ws_size,
                              hipStream_t stream) {
    const float* x  = (const float*)d_in[0];
    const float* Wq = (const float*)d_in[1];
    const float* Wk = (const float*)d_in[2];
    const float* Wv = (const float*)d_in[3];
    const float* Wo = (const float*)d_in[4];
    const float* bo = (const float*)d_in[5];

    const int M = B_BATCH * T_SEQ;   // 8192 rows
    const int C = C_DIM;             // 1024

    float* out  = (float*)d_out;                          // [B*T, C]
    float* attn = out + (size_t)M * C;                    // [B,H,T,T]

    const size_t nBTC = (size_t)M * C;                    // 8,388,608
    float* q = (float*)d_ws;                              // needs 4*32MB = 128MB
    float* k = q + nBTC;
    float* v = k + nBTC;
    float* y = v + nBTC;

    dim3 gemm_grid(C / TILE_N, M / TILE_M);               // (8, 64)
    dim3 blk(256);

    gemm_xt_wmma<<<gemm_grid, blk, 0, stream>>>(x, Wq, nullptr, q, M, C, C);
    gemm_xt_wmma<<<gemm_grid, blk, 0, stream>>>(x, Wk, nullptr, k, M, C, C);
    gemm_xt_wmma<<<gemm_grid, blk, 0, stream>>>(x, Wv, nullptr, v, M, C, C);

    const size_t attn_n4 = ((size_t)B_BATCH * H_NUM * T_SEQ * T_SEQ) / 4;
    fill_attn_zero<<<4096, blk, 0, stream>>>((v4f*)attn, attn_n4);

    const int n_waves = B_BATCH * H_NUM * T_SEQ;          // 131072
    local_attn<<<n_waves / 8, blk, 0, stream>>>(q, k, v, y, attn);

    gemm_xt_wmma<<<gemm_grid, blk, 0, stream>>>(y, Wo, bo, out, M, C, C);
}